// LMR_18296560680986
// MI455X (gfx1250) — compile-verified
//
#include <hip/hip_runtime.h>
#include <math.h>

#define Bn 4096
#define Dn 2048
#define Cn 64

typedef __attribute__((ext_vector_type(2))) float v2f;
typedef __attribute__((ext_vector_type(8))) float v8f;

// workspace layout (float offsets)
#define OFF_MINV    ((size_t)0)
#define OFF_RSTD    ((size_t)Bn)
#define OFF_CONTRIB ((size_t)2*Bn)
#define OFF_FSM     ((size_t)3*Bn)
#define OFF_LSUM    ((size_t)4*Bn)
#define OFF_E       ((size_t)8*Bn)
#define OFF_R       (OFF_E + (size_t)Bn*(size_t)Bn)

// ---------------------------------------------------------------------------
// Kernel 1: per-row stats (mean/var -> scale,bias), contrib/fs-mask, y_mix
// ---------------------------------------------------------------------------
__global__ void k_stats(const float* __restrict__ x, const int* __restrict__ y,
                        const float* __restrict__ alpha, const int* __restrict__ bidx,
                        const float* __restrict__ c, const int* __restrict__ fsc,
                        float* __restrict__ ws, float* __restrict__ ymix)
{
    int row = blockIdx.x;
    int tid = threadIdx.x;
    __shared__ float s1[256];
    __shared__ float s2[256];
    const float* xr = x + (size_t)row * Dn;
    float sum = 0.f, sq = 0.f;
    for (int j = tid; j < Dn; j += 256) { float v = xr[j]; sum += v; sq += v * v; }
    s1[tid] = sum; s2[tid] = sq;
    __syncthreads();
    for (int s = 128; s > 0; s >>= 1) {
        if (tid < s) { s1[tid] += s1[tid + s]; s2[tid] += s2[tid + s]; }
        __syncthreads();
    }
    if (tid == 0) {
        float mu  = s1[0] * (1.f / (float)Dn);
        float var = s2[0] * (1.f / (float)Dn) - mu * mu;
        float scale = rsqrtf(var + 1e-5f) * rsqrtf((float)Dn);
        ws[OFF_RSTD + row] = scale;
        ws[OFF_MINV + row] = -mu * scale;
        int yi = y[row];
        ws[OFF_CONTRIB + row] = c[yi];
        ws[OFF_FSM + row] = (float)fsc[yi];
    }
    if (tid < Cn) {
        int yi = y[row];
        int bi = bidx[row];
        int yb = y[bi];
        float a = alpha[row];
        float v = ((tid == yi) ? a : 0.f) + ((tid == yb) ? (1.f - a) : 0.f);
        ymix[(size_t)row * Cn + tid] = v;
    }
}

// ---------------------------------------------------------------------------
// Kernel 2: E = exp( mask( x_norm @ x_norm^T ) )   via V_WMMA_F32_16X16X4_F32
// block = 256 thr (8 waves) -> 64 rows x 128 cols; wave -> 16 rows x 64 cols
// ---------------------------------------------------------------------------
__global__ void k_gemm1(const float* __restrict__ x, float* __restrict__ ws)
{
    const float* minv = ws + OFF_MINV;
    const float* rstd = ws + OFF_RSTD;
    const float* fsm  = ws + OFF_FSM;
    float* Ebuf = ws + OFF_E;

    int tid  = threadIdx.x;
    int lane = tid & 31;
    int w    = tid >> 5;
    int rb = blockIdx.x * 64  + (w & 3) * 16;
    int cb = blockIdx.y * 128 + (w >> 2) * 64;
    int m    = lane & 15;
    int koff = (lane >> 4) << 1;

    int ra = rb + m;
    float sA = rstd[ra], bA = minv[ra];
    const float* pA = x + (size_t)ra * Dn + koff;

    const float* pB[4];
    float sB[4], bB[4];
#pragma unroll
    for (int t = 0; t < 4; ++t) {
        int jr = cb + t * 16 + m;
        sB[t] = rstd[jr]; bB[t] = minv[jr];
        pB[t] = x + (size_t)jr * Dn + koff;
    }

    v8f z = {0.f,0.f,0.f,0.f,0.f,0.f,0.f,0.f};
    v8f acc[4];
#pragma unroll
    for (int t = 0; t < 4; ++t) acc[t] = z;

#pragma unroll 4
    for (int k = 0; k < Dn; k += 4) {
        v2f xa = *(const v2f*)(pA + k);
        v2f a; a.x = xa.x * sA + bA; a.y = xa.y * sA + bA;
#pragma unroll
        for (int t = 0; t < 4; ++t) {
            v2f xb = *(const v2f*)(pB[t] + k);
            v2f b; b.x = xb.x * sB[t] + bB[t]; b.y = xb.y * sB[t] + bB[t];
            acc[t] = __builtin_amdgcn_wmma_f32_16x16x4_f32(
                false, a, false, b, (short)0, acc[t], false, false);
        }
    }

    // epilogue: mask (diag | fs-col) with -1e5 then exp (underflows to exact 0)
    int rowBase = rb + ((lane >> 4) << 3);
#pragma unroll
    for (int t = 0; t < 4; ++t) {
        int jcol = cb + t * 16 + m;
        float f = fsm[jcol];
#pragma unroll
        for (int v = 0; v < 8; ++v) {
            int row = rowBase + v;
            float val = acc[t][v];
            float msk = (row == jcol) ? 1.f : f;
            val -= 100000.f * msk;
            Ebuf[(size_t)row * Bn + jcol] = __expf(val);
        }
    }
}

// ---------------------------------------------------------------------------
// Kernel 3: softmax denominators  l[row] = sum_j E[row][j]
// ---------------------------------------------------------------------------
__global__ void k_rowsum(float* __restrict__ ws)
{
    int row = blockIdx.x;
    int tid = threadIdx.x;
    const float* Er = ws + OFF_E + (size_t)row * Bn;
    __shared__ float s1[256];
    float sum = 0.f;
    for (int j = tid; j < Bn; j += 256) sum += Er[j];
    s1[tid] = sum;
    __syncthreads();
    for (int s = 128; s > 0; s >>= 1) {
        if (tid < s) s1[tid] += s1[tid + s];
        __syncthreads();
    }
    if (tid == 0) ws[OFF_LSUM + row] = s1[0];
}

// ---------------------------------------------------------------------------
// Kernel 4: R = (E @ x) * (c_y/l) + x * (1 - c_y)     via WMMA f32 16x16x4
// block -> 64 rows x 128 dcols; wave -> 16 rows x 64 dcols
// ---------------------------------------------------------------------------
__global__ void k_gemm2(const float* __restrict__ x, float* __restrict__ ws)
{
    const float* Ebuf    = ws + OFF_E;
    const float* contrib = ws + OFF_CONTRIB;
    const float* lsum    = ws + OFF_LSUM;
    float* R = ws + OFF_R;

    int tid  = threadIdx.x;
    int lane = tid & 31;
    int w    = tid >> 5;
    int rb = blockIdx.x * 64  + (w & 3) * 16;
    int db = blockIdx.y * 128 + (w >> 2) * 64;
    int m    = lane & 15;
    int koff = (lane >> 4) << 1;

    int ra = rb + m;
    const float* pA = Ebuf + (size_t)ra * Bn + koff;

    const float* pB[4];
#pragma unroll
    for (int t = 0; t < 4; ++t)
        pB[t] = x + (size_t)koff * Dn + (db + t * 16 + m);

    v8f z = {0.f,0.f,0.f,0.f,0.f,0.f,0.f,0.f};
    v8f acc[4];
#pragma unroll
    for (int t = 0; t < 4; ++t) acc[t] = z;

#pragma unroll 4
    for (int k = 0; k < Bn; k += 4) {
        v2f a = *(const v2f*)(pA + k);
#pragma unroll
        for (int t = 0; t < 4; ++t) {
            v2f b; b.x = pB[t][0]; b.y = pB[t][Dn];
            pB[t] += 4 * Dn;
            acc[t] = __builtin_amdgcn_wmma_f32_16x16x4_f32(
                false, a, false, b, (short)0, acc[t], false, false);
        }
    }

    int rowBase = rb + ((lane >> 4) << 3);
#pragma unroll
    for (int t = 0; t < 4; ++t) {
        int col = db + t * 16 + m;
#pragma unroll
        for (int v = 0; v < 8; ++v) {
            int row = rowBase + v;
            float ct = contrib[row];
            float li = lsum[row];
            float xv = x[(size_t)row * Dn + col];
            R[(size_t)row * Dn + col] = acc[t][v] * (ct / li) + xv * (1.f - ct);
        }
    }
}

// ---------------------------------------------------------------------------
// Kernel 5: x_mix = alpha*R + (1-alpha)*R[beta_idx]   (float4 vectorized)
// ---------------------------------------------------------------------------
__global__ void k_mix(const float* __restrict__ alpha, const int* __restrict__ bidx,
                      const float* __restrict__ Rbuf, float* __restrict__ out)
{
    int idx = blockIdx.x * 256 + threadIdx.x;   // one float4 per thread
    int i  = idx >> 9;                          // Dn/4 = 512 groups per row
    int d4 = idx & 511;
    float a = alpha[i];
    int bi = bidx[i];
    const float4* R4 = (const float4*)Rbuf;
    float4 r1 = R4[(size_t)i  * 512 + d4];
    float4 r2 = R4[(size_t)bi * 512 + d4];
    float4 o;
    o.x = a * r1.x + (1.f - a) * r2.x;
    o.y = a * r1.y + (1.f - a) * r2.y;
    o.z = a * r1.z + (1.f - a) * r2.z;
    o.w = a * r1.w + (1.f - a) * r2.w;
    ((float4*)out)[idx] = o;
}

// ---------------------------------------------------------------------------
extern "C" void kernel_launch(void* const* d_in, const int* in_sizes, int n_in,
                              void* d_out, int out_size, void* d_ws, size_t ws_size,
                              hipStream_t stream)
{
    const float* x     = (const float*)d_in[0];
    const int*   y     = (const int*)  d_in[1];
    const float* alpha = (const float*)d_in[2];
    const int*   bidx  = (const int*)  d_in[3];
    const float* c     = (const float*)d_in[4];
    const int*   fsc   = (const int*)  d_in[5];
    float* out  = (float*)d_out;
    float* ws   = (float*)d_ws;
    float* ymix = out + (size_t)Bn * Dn;

    k_stats <<<Bn, 256, 0, stream>>>(x, y, alpha, bidx, c, fsc, ws, ymix);
    k_gemm1 <<<dim3(Bn / 64, Bn / 128), 256, 0, stream>>>(x, ws);
    k_rowsum<<<Bn, 256, 0, stream>>>(ws);
    k_gemm2 <<<dim3(Bn / 64, Dn / 128), 256, 0, stream>>>(x, ws);
    k_mix   <<<(Bn * (Dn / 4)) / 256, 256, 0, stream>>>(alpha, bidx, ws + OFF_R, out);
}